// VSampling_84052509982734
// MI455X (gfx1250) — compile-verified
//
#include <hip/hip_runtime.h>

// ---------------------------------------------------------------------------
// Voxel-mean sampling (pointops voxelize + segment mean) for gfx1250.
// Dense-grid histogram strategy: key space K = B*G^3 <= 16*100^3 < 2^24.
// Workspace layout (d_ws):
//   [0,1024)            Meta (per-batch min bits, gmax, G, G3, K, total)
//   cnt  : 2^24 int     per-cell point count
//   sx   : 2^24 float   per-cell sum x
//   sy   : 2^24 float   per-cell sum y
//   sz   : 2^24 float   per-cell sum z
//   part : 4096 int     per-scan-block partial sums (-> exclusive offsets)
// Output layout (d_out): n_p [N*3] f32 | n_o [B] i32 | counts [N] i32
// ---------------------------------------------------------------------------

#define VOXEL_SZ   0.05f
#define KMAXV      (1 << 24)           // dense key-space bound
#define SCAN_TILE  4096                // elements per scan block (256 thr * 16)
#define NBLK       (KMAXV / SCAN_TILE) // 4096 scan blocks
#define TPB        256                 // 8 waves (wave32)
#define PPB        1024                // points per scatter tile (12KB LDS)
#define SCATTER_BLOCKS 1024            // persistent blocks for the scatter pipeline

struct Meta {
  unsigned startBits[64 * 3];  // per-batch per-axis min coord as float bits (coords >= 0)
  int gmax;
  int G, G3, K, total;
};

__device__ __forceinline__ void fatomic_add(float* p, float v) {
  // lowers to global_atomic_add_f32 on gfx1250 (no CAS loop)
  unsafeAtomicAdd(p, v);
}

// Async global->LDS copy of 16 bytes (per-lane), tracked by ASYNCcnt (in-order).
__device__ __forceinline__ void async_copy_b128(unsigned lds_byte_addr,
                                                unsigned long long gaddr) {
  asm volatile("global_load_async_to_lds_b128 %0, %1, off"
               :
               : "v"(lds_byte_addr), "v"(gaddr)
               : "memory");
}

__device__ __forceinline__ void wait_asynccnt0() {
  asm volatile("s_wait_asynccnt 0x0" ::: "memory");
}
// ASYNCcnt completes in order: <=3 outstanding proves the previous tile's
// 3 copies landed while the next tile's 3 remain in flight.
__device__ __forceinline__ void wait_asynccnt3() {
  asm volatile("s_wait_asynccnt 0x3" ::: "memory");
}

__global__ void k_init_meta(Meta* __restrict__ m) {
  int t = threadIdx.x;
  if (t < 64 * 3) m->startBits[t] = 0x7F800000u;  // +inf bits
  if (t == 0) { m->gmax = 0; m->total = 0; }
}

// --- per-batch coordinate minimum (LDS-staged to avoid global contention) ---
__global__ __launch_bounds__(TPB) void k_min(const float* __restrict__ coord,
                                             const int* __restrict__ off,
                                             int N, int B, Meta* __restrict__ meta) {
  __shared__ unsigned umin[64 * 3];
  __shared__ int soff[64];
  int t = threadIdx.x;
  for (int j = t; j < 64 * 3; j += TPB) umin[j] = 0x7F800000u;
  if (t < B) soff[t] = off[t];
  __syncthreads();
  int i = blockIdx.x * TPB + t;
  if (i < N) {
    float x = coord[3 * i + 0], y = coord[3 * i + 1], z = coord[3 * i + 2];
    int b = 0;
    while (b < B - 1 && i >= soff[b]) ++b;   // searchsorted(offset, i, 'right')
    atomicMin(&umin[b * 3 + 0], __float_as_uint(x));
    atomicMin(&umin[b * 3 + 1], __float_as_uint(y));
    atomicMin(&umin[b * 3 + 2], __float_as_uint(z));
  }
  __syncthreads();
  for (int j = t; j < 3 * B; j += TPB)
    if (umin[j] != 0x7F800000u) atomicMin(&meta->startBits[j], umin[j]);
}

// --- global max grid index (-> G) ---
__global__ __launch_bounds__(TPB) void k_gmax(const float* __restrict__ coord,
                                              const int* __restrict__ off,
                                              int N, int B, Meta* __restrict__ meta) {
  __shared__ int soff[64];
  __shared__ int sh[TPB];
  int t = threadIdx.x;
  if (t < B) soff[t] = off[t];
  __syncthreads();
  int i = blockIdx.x * TPB + t;
  int m = 0;
  if (i < N) {
    float x = coord[3 * i + 0], y = coord[3 * i + 1], z = coord[3 * i + 2];
    int b = 0;
    while (b < B - 1 && i >= soff[b]) ++b;
    float bx = __uint_as_float(meta->startBits[b * 3 + 0]);
    float by = __uint_as_float(meta->startBits[b * 3 + 1]);
    float bz = __uint_as_float(meta->startBits[b * 3 + 2]);
    int gx = (int)floorf((x - bx) / VOXEL_SZ);
    int gy = (int)floorf((y - by) / VOXEL_SZ);
    int gz = (int)floorf((z - bz) / VOXEL_SZ);
    m = max(gx, max(gy, gz));
  }
  sh[t] = m;
  __syncthreads();
  for (int d = TPB / 2; d > 0; d >>= 1) {
    if (t < d) sh[t] = max(sh[t], sh[t + d]);
    __syncthreads();
  }
  if (t == 0) atomicMax(&meta->gmax, sh[0]);
}

__global__ void k_finalize(Meta* __restrict__ meta, int B) {
  int G = meta->gmax + 1;
  meta->G = G;
  meta->G3 = G * G * G;
  meta->K = B * G * G * G;   // <= 16*100^3 < 2^24, fits int
}

// --- scatter: persistent blocks, double-buffered async LDS staging ---------
__global__ __launch_bounds__(TPB) void k_scatter(const float* __restrict__ coord,
                                                 const int* __restrict__ off,
                                                 int N, int B, int ntiles,
                                                 const Meta* __restrict__ meta,
                                                 int* __restrict__ cnt,
                                                 float* __restrict__ sx,
                                                 float* __restrict__ sy,
                                                 float* __restrict__ sz) {
  __shared__ int soff[64];
  __shared__ float buf[2][3 * PPB];     // 2 x 12KB coord tiles
  int t = threadIdx.x;
  if (t < B) soff[t] = off[t];
  int G = meta->G;
  float start[64 * 3 > 0 ? 1 : 1];      // (placeholder to keep compiler quiet)
  (void)start;

  unsigned lds0 = (unsigned)(unsigned long long)(&buf[0][0]);
  unsigned lds1 = (unsigned)(unsigned long long)(&buf[1][0]);

  auto issue = [&](int tile, int p) {
    unsigned long long g0 =
        (unsigned long long)(const void*)(coord + 3 * (long)tile * PPB);
    unsigned base = p ? lds1 : lds0;
#pragma unroll
    for (int j = 0; j < 3; ++j) {
      unsigned e = (unsigned)(j * TPB + t) * 16u;
      async_copy_b128(base + e, g0 + e);
    }
  };
  auto tile_full = [&](int tile) {
    return (tile < ntiles) && (((long)tile + 1) * PPB <= (long)N);
  };

  int tile0 = blockIdx.x;
  int stride = gridDim.x;
  if (tile_full(tile0)) issue(tile0, 0);          // prologue: fill buffer 0

  int p = 0;
  for (int tile = tile0; tile < ntiles; tile += stride, p ^= 1) {
    long base = (long)tile * PPB;
    bool full = (base + PPB) <= (long)N;          // uniform per block
    int next = tile + stride;
    bool nfull = tile_full(next);
    if (nfull) issue(next, p ^ 1);                // overlap next copy with compute
    if (full) { if (nfull) wait_asynccnt3(); else wait_asynccnt0(); }
    __syncthreads();                              // publish buf[p] block-wide

    const float* bp = &buf[p][0];
#pragma unroll
    for (int j = 0; j < PPB / TPB; ++j) {
      int l = t * (PPB / TPB) + j;
      long i = base + l;
      if (i >= N) break;
      float x, y, z;
      if (full) {
        x = bp[3 * l + 0]; y = bp[3 * l + 1]; z = bp[3 * l + 2];
      } else {
        x = coord[3 * i + 0]; y = coord[3 * i + 1]; z = coord[3 * i + 2];
      }
      int b = 0;
      while (b < B - 1 && i >= (long)soff[b]) ++b;
      float bx = __uint_as_float(meta->startBits[b * 3 + 0]);
      float by = __uint_as_float(meta->startBits[b * 3 + 1]);
      float bz = __uint_as_float(meta->startBits[b * 3 + 2]);
      int gx = (int)floorf((x - bx) / VOXEL_SZ);
      int gy = (int)floorf((y - by) / VOXEL_SZ);
      int gz = (int)floorf((z - bz) / VOXEL_SZ);
      int key = ((b * G + gx) * G + gy) * G + gz;
      atomicAdd(&cnt[key], 1);
      fatomic_add(&sx[key], x);
      fatomic_add(&sy[key], y);
      fatomic_add(&sz[key], z);
    }
    __syncthreads();                              // all reads done before reuse
  }
}

// --- scan pass 1: per-block occupancy sums ---
__global__ __launch_bounds__(TPB) void k_pass1(const int* __restrict__ cnt,
                                               int* __restrict__ part) {
  __shared__ int sh[TPB];
  int t = threadIdx.x;
  int base = blockIdx.x * SCAN_TILE + t * 16;
  int s = 0;
  const int4* c4 = (const int4*)(cnt + base);
#pragma unroll
  for (int q = 0; q < 4; ++q) {
    int4 v = c4[q];
    s += (v.x > 0) + (v.y > 0) + (v.z > 0) + (v.w > 0);
  }
  sh[t] = s;
  __syncthreads();
  for (int d = TPB / 2; d > 0; d >>= 1) {
    if (t < d) sh[t] += sh[t + d];
    __syncthreads();
  }
  if (t == 0) part[blockIdx.x] = sh[0];
}

// --- scan pass 2: exclusive scan of 4096 block partials (single block) ---
__global__ __launch_bounds__(1024) void k_pass2(int* __restrict__ part,
                                                Meta* __restrict__ meta) {
  __shared__ int sh[1024];
  int t = threadIdx.x;
  int p0 = part[4 * t + 0], p1 = part[4 * t + 1], p2 = part[4 * t + 2], p3 = part[4 * t + 3];
  int s = p0 + p1 + p2 + p3;
  sh[t] = s;
  __syncthreads();
  for (int d = 1; d < 1024; d <<= 1) {
    int v = (t >= d) ? sh[t - d] : 0;
    __syncthreads();
    sh[t] += v;
    __syncthreads();
  }
  int excl = sh[t] - s;
  part[4 * t + 0] = excl;
  part[4 * t + 1] = excl + p0;
  part[4 * t + 2] = excl + p0 + p1;
  part[4 * t + 3] = excl + p0 + p1 + p2;
  if (t == 1023) meta->total = excl + s;
}

// --- scan pass 3: local exclusive scan fused with compaction + outputs ---
__global__ __launch_bounds__(TPB) void k_pass3(const int* __restrict__ cnt,
                                               const float* __restrict__ sx,
                                               const float* __restrict__ sy,
                                               const float* __restrict__ sz,
                                               const int* __restrict__ part,
                                               const Meta* __restrict__ meta,
                                               int B,
                                               float* __restrict__ out_np,
                                               int* __restrict__ out_no,
                                               int* __restrict__ out_cnt) {
  __shared__ int sh[TPB];
  int t = threadIdx.x;
  int base = blockIdx.x * SCAN_TILE + t * 16;
  int c[16];
  int s = 0;
  const int4* c4 = (const int4*)(cnt + base);
#pragma unroll
  for (int q = 0; q < 4; ++q) {
    int4 v = c4[q];
    c[4 * q + 0] = v.x; c[4 * q + 1] = v.y; c[4 * q + 2] = v.z; c[4 * q + 3] = v.w;
    s += (v.x > 0) + (v.y > 0) + (v.z > 0) + (v.w > 0);
  }
  sh[t] = s;
  __syncthreads();
  for (int d = 1; d < TPB; d <<= 1) {
    int v = (t >= d) ? sh[t - d] : 0;
    __syncthreads();
    sh[t] += v;
    __syncthreads();
  }
  int run = part[blockIdx.x] + (sh[t] - s);  // global exclusive scan at `base`
  int G3 = meta->G3;
#pragma unroll
  for (int j = 0; j < 16; ++j) {
    int k = base + j;
    int cj = c[j];
    if (cj > 0) {
      float inv = 1.0f / (float)cj;
      out_np[3 * run + 0] = sx[k] * inv;
      out_np[3 * run + 1] = sy[k] * inv;
      out_np[3 * run + 2] = sz[k] * inv;
      out_cnt[run] = cj;
      ++run;                                // run == inclusive scan at k
    }
    int kp1 = k + 1;
    if ((kp1 % G3) == 0) {                  // batch boundary in key space
      int b2 = kp1 / G3 - 1;
      if (b2 >= 0 && b2 < B) out_no[b2] = run;  // cumulative voxel count (n_o)
    }
  }
}

extern "C" void kernel_launch(void* const* d_in, const int* in_sizes, int n_in,
                              void* d_out, int out_size, void* d_ws, size_t ws_size,
                              hipStream_t stream) {
  const float* coord = (const float*)d_in[0];
  const int*   off   = (const int*)d_in[1];
  int N = in_sizes[0] / 3;
  int B = in_sizes[1];

  Meta*  meta = (Meta*)d_ws;
  int*   cnt  = (int*)((char*)d_ws + 1024);
  float* sx   = (float*)(cnt + (size_t)KMAXV);
  float* sy   = sx + (size_t)KMAXV;
  float* sz   = sy + (size_t)KMAXV;
  int*   part = (int*)(sz + (size_t)KMAXV);

  float* out_np  = (float*)d_out;
  int*   out_no  = (int*)(out_np + (size_t)3 * N);
  int*   out_cnt = out_no + B;

  // Re-initialize everything we read (deterministic across graph replays).
  hipMemsetAsync(cnt, 0, (size_t)KMAXV * 4 * sizeof(int), stream);    // cnt+sx+sy+sz
  hipMemsetAsync(d_out, 0, (size_t)out_size * sizeof(float), stream); // zero padding
  k_init_meta<<<1, 256, 0, stream>>>(meta);

  int nb = (N + TPB - 1) / TPB;
  k_min<<<nb, TPB, 0, stream>>>(coord, off, N, B, meta);
  k_gmax<<<nb, TPB, 0, stream>>>(coord, off, N, B, meta);
  k_finalize<<<1, 1, 0, stream>>>(meta, B);

  int ntiles = (int)(((long)N + PPB - 1) / PPB);
  int nscatter = ntiles < SCATTER_BLOCKS ? ntiles : SCATTER_BLOCKS;
  k_scatter<<<nscatter, TPB, 0, stream>>>(coord, off, N, B, ntiles, meta,
                                          cnt, sx, sy, sz);

  k_pass1<<<NBLK, TPB, 0, stream>>>(cnt, part);
  k_pass2<<<1, 1024, 0, stream>>>(part, meta);
  k_pass3<<<NBLK, TPB, 0, stream>>>(cnt, sx, sy, sz, part, meta, B,
                                    out_np, out_no, out_cnt);
}